// GCN_18700287607644
// MI455X (gfx1250) — compile-verified
//
#include <hip/hip_runtime.h>
#include <hip/hip_bf16.h>
#include <math.h>

// ---------------------------------------------------------------------------
// GCN 2-layer forward for MI455X (gfx1250, wave32).
// Dense GEMMs: V_WMMA_F32_16X16X4_F32. B staged in LDS in K-pair-interleaved
// float2 layout so each B fragment is a single aligned ds_load_b64 (no VGPR
// repacking). Each wave computes a 16 x (NT*16) strip with NT independent
// accumulators (A-fragment reuse, independent WMMA chains).
// Sparse aggregation: wave-per-edge with global_atomic_add_f32.
// ---------------------------------------------------------------------------

typedef float v2f __attribute__((ext_vector_type(2)));
typedef float v8f __attribute__((ext_vector_type(8)));

// ---------------- init / elementwise helpers ----------------

__global__ void set_const_kernel(float* __restrict__ p, size_t n, float v) {
  size_t i = (size_t)blockIdx.x * blockDim.x + threadIdx.x;
  size_t stride = (size_t)gridDim.x * blockDim.x;
  for (; i < n; i += stride) p[i] = v;
}

__global__ void degree_kernel(const int* __restrict__ cols, float* __restrict__ deg, int E) {
  int i = blockIdx.x * blockDim.x + threadIdx.x;
  if (i < E) atomicAdd(&deg[cols[i]], 1.0f);
}

__global__ void rsqrt_kernel(float* __restrict__ deg, int n) {
  int i = blockIdx.x * blockDim.x + threadIdx.x;
  if (i < n) deg[i] = rsqrtf(deg[i]);  // deg >= 1 always (self-loop)
}

__global__ void bias_relu_kernel(float* __restrict__ a, const float* __restrict__ b,
                                 size_t n, int F) {
  size_t i = (size_t)blockIdx.x * blockDim.x + threadIdx.x;
  if (i < n) {
    int f = (int)(i % (size_t)F);
    a[i] = fmaxf(a[i] + b[f], 0.0f);
  }
}

// ---------------- WMMA fp32 GEMM: C[M x Nvalid] = A[M x K] @ B[K x Nvalid] ----
// K, NT compile-time. Npad = NT*16 >= Nvalid. Block = 8 waves.
// LDS layout: bs[kk*NP + n] = { B[2kk][n], B[2kk+1][n] }  (K-pair interleave)
// -> the B fragment for WMMA lane (khalf = (lane>>4)*2) at K-step k is the
//    single float2 at bs[(k/2 + (lane>>4))*NP + (lane&15) + t*16]:
//    one aligned ds_load_b64 straight into the operand VGPR pair.
//
// A (16x4 f32, 2 VGPRs): lane = (k>=2)*16 + m ; vgpr = k&1
// B (4x16 f32, 2 VGPRs): lane = (k>=2)*16 + n ; vgpr = k&1
// C (16x16 f32, 8 VGPRs): vgpr v, lanes 0-15 -> M=v, lanes 16-31 -> M=v+8

template <int K, int NT>
__global__ void gemm_wmma_f32(const float* __restrict__ A, const float* __restrict__ B,
                              float* __restrict__ C, int M, int Nvalid) {
  constexpr int NP = NT * 16;
  constexpr int KP = K / 2;       // number of K-pairs
  __shared__ v2f bs[KP * NP];

  // cooperative, one-shot staging of B into LDS (zero pad columns >= Nvalid)
  const int tid = threadIdx.y * 32 + threadIdx.x;
  for (int i = tid; i < KP * NP; i += 256) {
    const int kk = i / NP, n = i % NP;
    v2f v = {0.0f, 0.0f};
    if (n < Nvalid) {
      v.x = B[(size_t)(2 * kk) * Nvalid + n];
      v.y = B[(size_t)(2 * kk + 1) * Nvalid + n];
    }
    bs[i] = v;
  }
  __syncthreads();

  const int lane = threadIdx.x & 31;
  const int tm   = blockIdx.x * blockDim.y + threadIdx.y;  // wave-uniform M tile
  if (tm * 16 >= M) return;                                // wave-uniform exit

  const int mrow  = tm * 16 + (lane & 15);
  const int nlo   = lane & 15;
  const int krow  = lane >> 4;        // 0 for lanes 0-15, 1 for lanes 16-31
  const int khalf = krow << 1;        // 0 or 2

  v8f acc[NT];
#pragma unroll
  for (int t = 0; t < NT; ++t) acc[t] = (v8f){0, 0, 0, 0, 0, 0, 0, 0};

  const float* ap = A + (size_t)mrow * K + khalf;
#pragma unroll
  for (int k2 = 0; k2 < KP; k2 += 2) {   // K-step of 4 = 2 pairs
    v2f a;
    a.x = ap[2 * k2];
    a.y = ap[2 * k2 + 1];
    const v2f* bp = &bs[(k2 + krow) * NP + nlo];
#pragma unroll
    for (int t = 0; t < NT; ++t) {
      v2f b = bp[t * 16];  // single ds_load_b64, already {row k, row k+1}
      acc[t] = __builtin_amdgcn_wmma_f32_16x16x4_f32(
          /*neg_a=*/false, a, /*neg_b=*/false, b,
          /*c_mod=*/(short)0, acc[t], /*reuse_a=*/false, /*reuse_b=*/false);
    }
  }

  const int mbase = tm * 16 + (krow << 3);
#pragma unroll
  for (int t = 0; t < NT; ++t) {
    const int ncol = t * 16 + nlo;
    if (ncol < Nvalid) {
#pragma unroll
      for (int v = 0; v < 8; ++v)
        C[(size_t)(mbase + v) * Nvalid + ncol] = acc[t][v];
    }
  }
}

// ---------------- sparse aggregation: out[c] += dinv[r]*dinv[c] * h[r] --------
// One wave per edge; edges [E, E+Nnodes) are the implicit self-loops.

__global__ void scatter_kernel(const int* __restrict__ rows, const int* __restrict__ cols,
                               const float* __restrict__ dinv, const float* __restrict__ h,
                               float* __restrict__ out, int E, int Nnodes, int F) {
  const int wave = blockIdx.x * (blockDim.x >> 5) + (threadIdx.x >> 5);
  const int lane = threadIdx.x & 31;
  const int Etot = E + Nnodes;
  if (wave >= Etot) return;
  int r, c;
  if (wave < E) { r = rows[wave]; c = cols[wave]; }
  else          { r = c = wave - E; }
  const float norm = dinv[r] * dinv[c];
  const float* hp = h + (size_t)r * F;
  float* op = out + (size_t)c * F;
  for (int f = lane; f < F; f += 32)
    atomicAdd(&op[f], norm * hp[f]);
}

// ---------------- row-wise log-softmax (wave per row, wave32 shuffles) --------

__global__ void logsoftmax_kernel(const float* __restrict__ agg, const float* __restrict__ bias,
                                  float* __restrict__ out, int Nnodes, int F) {
  const int row  = blockIdx.x * (blockDim.x >> 5) + (threadIdx.x >> 5);
  const int lane = threadIdx.x & 31;
  if (row >= Nnodes) return;
  const float* p = agg + (size_t)row * F;
  const float ninf = -__builtin_inff();
  float v0 = (lane < F)      ? p[lane] + bias[lane]           : ninf;
  float v1 = (lane + 32 < F) ? p[lane + 32] + bias[lane + 32] : ninf;
  float m = fmaxf(v0, v1);
#pragma unroll
  for (int off = 16; off > 0; off >>= 1) m = fmaxf(m, __shfl_xor(m, off, 32));
  float s = __expf(v0 - m) + __expf(v1 - m);  // exp(-inf)=0 handles padding
#pragma unroll
  for (int off = 16; off > 0; off >>= 1) s += __shfl_xor(s, off, 32);
  const float lse = m + __logf(s);
  float* o = out + (size_t)row * F;
  if (lane < F)      o[lane]      = v0 - lse;
  if (lane + 32 < F) o[lane + 32] = v1 - lse;
}

// ---------------------------------------------------------------------------

extern "C" void kernel_launch(void* const* d_in, const int* in_sizes, int n_in,
                              void* d_out, int out_size, void* d_ws, size_t ws_size,
                              hipStream_t stream) {
  const float* x  = (const float*)d_in[0];   // [N, 128]
  const int*   ei = (const int*)d_in[1];     // [2, E] flat: rows then cols
  const float* W1 = (const float*)d_in[2];   // [128, 64]
  const float* b1 = (const float*)d_in[3];   // [64]
  const float* W2 = (const float*)d_in[4];   // [64, 40]
  const float* b2 = (const float*)d_in[5];   // [40]
  float* out = (float*)d_out;                // [N, 40] log-softmax

  const int Nn = in_sizes[0] / 128;   // 50000 (multiple of 16)
  const int E  = in_sizes[1] / 2;     // 800000
  const int F1 = 64, F2 = 40;

  // workspace layout (floats)
  float* deg  = (float*)d_ws;                 // [Nn]    degree -> dinv (in place)
  float* h1   = deg + Nn;                     // [Nn*64] x@W1
  float* a1   = h1 + (size_t)Nn * F1;         // [Nn*64] aggregated + relu
  float* h2   = a1 + (size_t)Nn * F1;         // [Nn*40] a1@W2
  float* agg2 = h2 + (size_t)Nn * F2;         // [Nn*40] aggregated layer 2

  const int* rows = ei;
  const int* cols = ei + E;

  // --- init (ws is poisoned by harness; deterministic re-init every call) ---
  set_const_kernel<<<512, 256, 0, stream>>>(deg, (size_t)Nn, 1.0f);       // self-loop
  set_const_kernel<<<2048, 256, 0, stream>>>(a1, (size_t)Nn * F1, 0.0f);
  set_const_kernel<<<2048, 256, 0, stream>>>(agg2, (size_t)Nn * F2, 0.0f);

  // --- symmetric normalization: dinv = rsqrt(1 + in_degree) ---
  degree_kernel<<<(E + 255) / 256, 256, 0, stream>>>(cols, deg, E);
  rsqrt_kernel<<<(Nn + 255) / 256, 256, 0, stream>>>(deg, Nn);

  const int mtiles = (Nn + 15) / 16;  // 3125
  const dim3 blk(32, 8);              // 8 waves = 8 M-tiles per block

  // --- layer 1: h1 = x @ W1 (WMMA, K=128, Npad=64), aggregate, bias+relu ---
  gemm_wmma_f32<128, 4><<<(mtiles + 7) / 8, blk, 0, stream>>>(x, W1, h1, Nn, F1);
  {
    const int Etot = E + Nn;  // wave per edge, 8 waves per block
    scatter_kernel<<<(Etot + 7) / 8, 256, 0, stream>>>(rows, cols, deg, h1, a1, E, Nn, F1);
  }
  {
    const size_t n = (size_t)Nn * F1;
    bias_relu_kernel<<<(unsigned)((n + 255) / 256), 256, 0, stream>>>(a1, b1, n, F1);
  }

  // --- layer 2: h2 = a1 @ W2 (WMMA, K=64, Npad=48), aggregate, log-softmax ---
  gemm_wmma_f32<64, 3><<<(mtiles + 7) / 8, blk, 0, stream>>>(a1, W2, h2, Nn, F2);
  {
    const int Etot = E + Nn;
    scatter_kernel<<<(Etot + 7) / 8, 256, 0, stream>>>(rows, cols, deg, h2, agg2, E, Nn, F2);
  }
  logsoftmax_kernel<<<(Nn + 7) / 8, 256, 0, stream>>>(agg2, b2, out, Nn, F2);
}